// DiceLoss_91147795955894
// MI455X (gfx1250) — compile-verified
//
#include <hip/hip_runtime.h>
#include <cstdint>
#include <cstddef>

// ---------------- problem constants ----------------
#define C_CLASSES 1328
#define HW_PIX    65536           // 256*256
#define NPIX      131072          // B*H*W = 2*65536
#define TILE_PIX  16
#define N_TILES   (NPIX / TILE_PIX)      // 8192
#define NBLK      512
#define TILES_PER_BLK (N_TILES / NBLK)   // 16
#define ROWS      1344            // C padded to multiple of 4 (b64 DMA packs 4 rows/instr)
#define DMA_INSTR (ROWS / 4)      // 336 b64 instructions per tile
#define RSTR      18              // floats per LDS row: 16 px + 2 pad (conflict-free col reads)
#define LOG2E_F   1.4426950408889634f
#define IGNORE_IDX 255

static_assert(N_TILES % NBLK == 0, "tiles must divide blocks");
static_assert((ROWS % 4) == 0, "b64 DMA packs 4 rows per instruction");
static_assert((DMA_INSTR % 8) == 0, "uniform DMA split across 8 waves");

// ---------------- CDNA5 async-to-LDS ----------------
#if defined(__has_builtin)
#  if __has_builtin(__builtin_amdgcn_global_load_async_to_lds_b64)
#    define HAVE_ASYNC_LDS 1
#  endif
#  if __has_builtin(__builtin_amdgcn_s_wait_asynccnt)
#    define HAVE_WAIT_ASYNC_BUILTIN 1
#  endif
#endif

#ifndef HAVE_ASYNC_LDS
#warning "CDNA5 probe: __builtin_amdgcn_global_load_async_to_lds_b64 NOT available; using sync LDS-staging fallback"
#endif

typedef int v2i __attribute__((ext_vector_type(2)));
typedef __attribute__((address_space(1))) v2i GV2i;   // global int2
typedef __attribute__((address_space(3))) v2i LV2i;   // LDS int2

__device__ __forceinline__ void async_load_b64(const float* g, float* lds) {
#ifdef HAVE_ASYNC_LDS
  __builtin_amdgcn_global_load_async_to_lds_b64((GV2i*)g, (LV2i*)lds, 0, 0);
#else
  // synchronous fallback: coalesced 8B global load -> LDS store
  lds[0] = g[0];
  lds[1] = g[1];
#endif
}

__device__ __forceinline__ void wait_async_all() {
#ifdef HAVE_ASYNC_LDS
#  ifdef HAVE_WAIT_ASYNC_BUILTIN
  __builtin_amdgcn_s_wait_asynccnt(0);
#  else
  asm volatile("s_wait_asynccnt 0" ::: "memory");
#  endif
#endif
}

// Issue async DMA of one 16-pixel tile (ROWS x 16 f32) into an LDS slab.
// b64 = 8B/lane -> 32 lanes cover 4 rows (4 classes) per instruction.
// instruction k = w + 8*u, k < DMA_INSTR; lane l: row = 4k + l/8, px = 2*(l%8).
__device__ __forceinline__ void dma_issue_tile(const float* __restrict__ logits,
                                               int pixBase, float* __restrict__ slab,
                                               int w, int l) {
  const int b   = pixBase >> 16;        // HW_PIX = 65536
  const int hw0 = pixBase & 0xFFFF;
  const float* gbase = logits + (size_t)b * C_CLASSES * HW_PIX + hw0;
  const int rsub = l >> 3;              // 0..3
  const int px   = (l & 7) * 2;         // 0,2,..,14
#pragma unroll 4
  for (int u = 0; u < DMA_INSTR / 8; ++u) {          // 42 instrs per wave, uniform
    const int c  = 4 * (w + 8 * u) + rsub;           // LDS row 0..1343
    const int cg = (c < C_CLASSES) ? c : (C_CLASSES - 1);  // clamp global read (no OOB)
    const float* g = gbase + (size_t)cg * HW_PIX + px;
    float* d = slab + c * RSTR + px;                 // 8B-aligned (c*18 even, px even)
    async_load_b64(g, d);
  }
}

// ---------------- main kernel (first in module so disasm snippet shows it) ----------------
__global__ __launch_bounds__(256)
void dice_main_kernel(const float* __restrict__ logits,
                      const int* __restrict__ labels,
                      float* __restrict__ g_inter,
                      float* __restrict__ g_count,
                      float* __restrict__ g_psum) {
  __shared__ float slab[2][ROWS * RSTR];       // 2 x 96,768 B double buffer
  __shared__ float mP[256], zP[256];           // per-thread softmax partials
  __shared__ float sArr[TILE_PIX];             // s = max + log2(sumexp2) per pixel
  __shared__ int   labArr[TILE_PIX];
  __shared__ float validArr[TILE_PIX];
  __shared__ float probAcc[C_CLASSES];         // block-local per-class prob sums

  const int t = threadIdx.x;
  const int w = t >> 5;          // wave id 0..7
  const int l = t & 31;          // lane
  const int p = l & 15;          // pixel within tile (sweep1 mapping)
  const int r = l >> 4;          // class-interleave half (sweep1 mapping)

  const int tile0 = blockIdx.x * TILES_PER_BLK;

  // prime the pipeline: DMA tile 0 into buffer 0 (issued before anything else)
  dma_issue_tile(logits, tile0 * TILE_PIX, &slab[0][0], w, l);

  for (int c = t; c < C_CLASSES; c += 256) probAcc[c] = 0.0f;

  for (int j = 0; j < TILES_PER_BLK; ++j) {
    const int buf = j & 1;
    const int pixBase = (tile0 + j) * TILE_PIX;

    // per-tile labels / validity (written before the barrier that publishes the slab)
    if (t < TILE_PIX) {
      const int li = labels[pixBase + t];
      labArr[t]   = li;
      validArr[t] = (li != IGNORE_IDX) ? 1.0f : 0.0f;
    }
    wait_async_all();       // my wave's async DMA for this buffer done (ASYNCcnt)
    __syncthreads();        // everyone's DMA done; labels visible

    // issue next tile's DMA NOW: buf^1 was fully consumed before the barrier above,
    // so the whole of this tile's compute hides the next tile's HBM transfer.
    if (j + 1 < TILES_PER_BLK)
      dma_issue_tile(logits, pixBase + TILE_PIX, &slab[buf ^ 1][0], w, l);

    // -------- sweep 1: online max / sum-exp2 over this wave's 166-class strip --------
    // thread covers classes c = w*166 + 2*i + r, pixel p.  base-2 domain (native v_exp_f32).
    {
      float m = -3.402823466e38f, Z = 0.0f;
      const int cbase = w * 166 + r;
      const float* __restrict__ sb = &slab[buf][0];
      for (int i = 0; i < 83; ++i) {
        const int c = cbase + 2 * i;
        const float u = sb[c * RSTR + p] * LOG2E_F;
        const float nm = fmaxf(m, u);
        Z = Z * __builtin_amdgcn_exp2f(m - nm) + __builtin_amdgcn_exp2f(u - nm);
        m = nm;
      }
      mP[t] = m;
      zP[t] = Z;
    }
    __syncthreads();

    // -------- merge 16 partials per pixel; intersection/count atomics --------
    if (t < TILE_PIX) {
      float mm = -3.402823466e38f;
      for (int k = 0; k < 16; ++k) mm = fmaxf(mm, mP[k * 16 + t]);
      float zz = 0.0f;
      for (int k = 0; k < 16; ++k)
        zz += zP[k * 16 + t] * __builtin_amdgcn_exp2f(mP[k * 16 + t] - mm);
      const float s = mm + __builtin_amdgcn_logf(zz);   // v_log_f32 is log2
      sArr[t] = s;
      if (validArr[t] > 0.0f) {
        // prob at label = direct LDS lookup into the resident slab (no in-loop capture)
        const float xl = slab[buf][labArr[t] * RSTR + t] * LOG2E_F;
        const float pl = __builtin_amdgcn_exp2f(xl - s);
        atomicAdd(&g_inter[labArr[t]], pl);
        atomicAdd(&g_count[labArr[t]], 1.0f);
      }
    }
    __syncthreads();

    // -------- sweep 2: per-class masked prob sums; lanes = classes --------
    // chunk k2 = w + 8*u2, classes c = 32*k2 + l  (disjoint per wave -> plain LDS RMW)
    for (int u2 = 0; u2 < 6; ++u2) {
      const int k2 = w + 8 * u2;
      if (k2 * 32 < C_CLASSES) {
        const int c  = k2 * 32 + l;
        const int cc = (c < C_CLASSES) ? c : (C_CLASSES - 1);
        const float* __restrict__ row = &slab[buf][cc * RSTR];
        float acc = 0.0f;
#pragma unroll
        for (int pp = 0; pp < TILE_PIX; ++pp) {
          const float e = __builtin_amdgcn_exp2f(row[pp] * LOG2E_F - sArr[pp]);
          acc += validArr[pp] * e;
        }
        if (c < C_CLASSES) probAcc[c] += acc;
      }
    }
    __syncthreads();  // protect labArr/validArr/sArr/slab before next tile rewrites them
  }

  // flush block-local per-class sums once
  for (int c = t; c < C_CLASSES; c += 256) atomicAdd(&g_psum[c], probAcc[c]);
}

// ---------------- epilogue kernels ----------------
__global__ void dice_zero_kernel(float* __restrict__ ws) {
  const int i = blockIdx.x * 256 + threadIdx.x;
  if (i < 3 * C_CLASSES) ws[i] = 0.0f;
}

__global__ void dice_final_kernel(const float* __restrict__ g_inter,
                                  const float* __restrict__ g_count,
                                  const float* __restrict__ g_psum,
                                  float* __restrict__ out) {
  __shared__ float sd[256], sp[256];
  const int t = threadIdx.x;
  float dsum = 0.0f, pres = 0.0f;
  for (int c = t; c < C_CLASSES; c += 256) {
    const float I = g_inter[c];
    const float K = g_count[c];
    const float P = g_psum[c];
    const float dice = (2.0f * I + 1.0f) / (P + K + 1.0f);
    if (K > 0.0f) { dsum += dice; pres += 1.0f; }
  }
  sd[t] = dsum; sp[t] = pres;
  __syncthreads();
  for (int st = 128; st > 0; st >>= 1) {
    if (t < st) { sd[t] += sd[t + st]; sp[t] += sp[t + st]; }
    __syncthreads();
  }
  if (t == 0) out[0] = (sp[0] > 0.0f) ? (1.0f - sd[0] / sp[0]) : 0.0f;
}

// ---------------- host entry ----------------
extern "C" void kernel_launch(void* const* d_in, const int* in_sizes, int n_in,
                              void* d_out, int out_size, void* d_ws, size_t ws_size,
                              hipStream_t stream) {
  const float* logits = (const float*)d_in[0];
  const int*   labels = (const int*)d_in[1];
  float* out = (float*)d_out;
  float* ws  = (float*)d_ws;            // needs 3*1328*4 = ~16 KB
  float* g_inter = ws;
  float* g_count = ws + C_CLASSES;
  float* g_psum  = ws + 2 * C_CLASSES;

  dice_zero_kernel<<<(3 * C_CLASSES + 255) / 256, 256, 0, stream>>>(ws);
  dice_main_kernel<<<NBLK, 256, 0, stream>>>(logits, labels, g_inter, g_count, g_psum);
  dice_final_kernel<<<1, 256, 0, stream>>>(g_inter, g_count, g_psum, out);
}